// EntropyLoss_84542136254557
// MI455X (gfx1250) — compile-verified
//
#include <hip/hip_runtime.h>
#include <hip/hip_bf16.h>
#include <math.h>

#define BB 4
#define KK 512
#define NN 8192

typedef float v2f __attribute__((ext_vector_type(2)));
typedef float v8f __attribute__((ext_vector_type(8)));

// ---------------------------------------------------------------------------
// Kernel 1: src_corr = R*src + t.
// Pack A rows as (-2*c0, -2*c1, -2*c2, 1.0)  -> srcT4
// Pack B cols as ( t_x,  t_y,  t_z,  |t|^2 ) -> tgtT4
// so the 16x16x4 WMMA directly yields  -2*<s,t> + yy[m].
// xx = |src_corr|^2 kept separately (row-constant, added after the min).
// ---------------------------------------------------------------------------
__global__ void __launch_bounds__(256)
prep_kernel(const float* __restrict__ src, const float* __restrict__ tgt,
            const float* __restrict__ rot, const float* __restrict__ trans,
            float* __restrict__ srcT4, float* __restrict__ tgtT4,
            float* __restrict__ xx) {
    int p = blockIdx.x * blockDim.x + threadIdx.x;   // 0 .. BB*NN-1 exactly
    int b = p / NN;
    int n = p - b * NN;
    const float* Rb = rot + b * 9;
    const float* tb = trans + b * 3;

    float sx = src[(b * 3 + 0) * NN + n];
    float sy = src[(b * 3 + 1) * NN + n];
    float sz = src[(b * 3 + 2) * NN + n];
    float c0 = Rb[0] * sx + Rb[1] * sy + Rb[2] * sz + tb[0];
    float c1 = Rb[3] * sx + Rb[4] * sy + Rb[5] * sz + tb[1];
    float c2 = Rb[6] * sx + Rb[7] * sy + Rb[8] * sz + tb[2];
    ((float4*)srcT4)[p] = make_float4(-2.0f * c0, -2.0f * c1, -2.0f * c2, 1.0f);
    xx[p] = c0 * c0 + c1 * c1 + c2 * c2;

    float tx = tgt[(b * 3 + 0) * NN + n];
    float ty = tgt[(b * 3 + 1) * NN + n];
    float tz = tgt[(b * 3 + 2) * NN + n];
    float yv = tx * tx + ty * ty + tz * tz;
    ((float4*)tgtT4)[p] = make_float4(tx, ty, tz, yv);
}

// ---------------------------------------------------------------------------
// Kernel 2: per-wave 16-row src tile; V_WMMA_F32_16X16X4_F32 against all tgt
// tiles.  D already equals the per-column candidate (-2*inner + yy), so the
// inner loop is just: load B float2 -> wmma -> 8x v_min.
// Prefetch uses locality 3 -> WGP scope (pull into all cache levels): the
// B-tile stream is shared by all 8 waves of the block.  Prefetching one tile
// past the end is safe (speculative prefetch is silently dropped).
// A layout (ISA 7.12.2, 32-bit A 16x4): lane l holds row (l&15),
// K pair = (l<16)?{0,1}:{2,3}.  B 4x16 uses the transposed-identical gather.
// D layout: lane l, vgpr r -> row r + (l<16?0:8), col l&15.
// ---------------------------------------------------------------------------
__global__ void __launch_bounds__(256)
nn_kernel(const float* __restrict__ srcT4, const float* __restrict__ tgtT4,
          const float* __restrict__ xx, float* __restrict__ nearst) {
    const int lane = threadIdx.x & 31;
    const int wave = threadIdx.x >> 5;
    const int tile = blockIdx.x * 8 + wave;        // 0 .. BB*(NN/16)-1
    const int b    = tile / (NN / 16);
    const int n0   = (tile - b * (NN / 16)) * 16;
    const int half = lane >> 4;                    // 0 or 1
    const int l15  = lane & 15;
    const int koff = half * 2;                     // K pair {0,1} or {2,3}

    const float* sb = srcT4 + (size_t)b * NN * 4;
    const float* tb = tgtT4 + (size_t)b * NN * 4;

    // A tile (constant across the m loop)
    v2f a = *(const v2f*)(sb + (size_t)(n0 + l15) * 4 + koff);

    v8f minv;
#pragma unroll
    for (int r = 0; r < 8; ++r) minv[r] = 3.0e38f;

    for (int m0 = 0; m0 < NN; m0 += 16) {
        // unguarded speculative prefetch of the next tile, WGP scope
        __builtin_prefetch(tb + (size_t)(m0 + 16 + l15) * 4 + koff, 0, 3);
        v2f bt = *(const v2f*)(tb + (size_t)(m0 + l15) * 4 + koff);
        v8f c = {};
        v8f d = __builtin_amdgcn_wmma_f32_16x16x4_f32(
            /*neg_a=*/false, a, /*neg_b=*/false, bt,
            /*c_mod=*/(short)0, c, /*reuse_a=*/false, /*reuse_b=*/false);
#pragma unroll
        for (int r = 0; r < 8; ++r) minv[r] = fminf(minv[r], d[r]);
    }

    // Min across the 16 lanes of each half (xor masks 1..8 stay in-half).
#pragma unroll
    for (int m = 1; m <= 8; m <<= 1) {
#pragma unroll
        for (int r = 0; r < 8; ++r)
            minv[r] = fminf(minv[r], __shfl_xor(minv[r], m, 32));
    }

    if (l15 == 0) {
        const int   base = n0 + half * 8;
        const float* xxb = xx     + (size_t)b * NN;
        float*       nb  = nearst + (size_t)b * NN;
#pragma unroll
        for (int r = 0; r < 8; ++r) nb[base + r] = xxb[base + r] + minv[r];
    }
}

// ---------------------------------------------------------------------------
// Kernel 3: per-batch bitonic sort of 8192 (dist, idx) pairs in LDS (64 KB),
// composite key (dist, idx) ascending == stable argsort.  Emit first K idx.
// ---------------------------------------------------------------------------
__global__ void __launch_bounds__(1024)
topk_sort_kernel(const float* __restrict__ nearst, int* __restrict__ topk) {
    __shared__ float sd[NN];
    __shared__ int   si[NN];
    const int b   = blockIdx.x;
    const int tid = threadIdx.x;

    for (int i = tid; i < NN; i += 1024) {
        sd[i] = nearst[(size_t)b * NN + i];
        si[i] = i;
    }
    __syncthreads();

    for (int k = 2; k <= NN; k <<= 1) {
        for (int j = k >> 1; j > 0; j >>= 1) {
            for (int i = tid; i < NN; i += 1024) {
                int l = i ^ j;
                if (l > i) {
                    bool up = ((i & k) == 0);
                    float di = sd[i], dl = sd[l];
                    int   xi = si[i], xl = si[l];
                    bool  gt = (di > dl) || (di == dl && xi > xl);
                    if (gt == up) {
                        sd[i] = dl; sd[l] = di;
                        si[i] = xl; si[l] = xi;
                    }
                }
            }
            __syncthreads();
        }
    }

    for (int i = tid; i < KK; i += 1024) topk[b * KK + i] = si[i];
}

// ---------------------------------------------------------------------------
// Kernel 4: loss = -(1/(B*K)) * sum_{b,j} log(scores[b, j, topk[b][j]])
// ---------------------------------------------------------------------------
__global__ void __launch_bounds__(1024)
loss_kernel(const float* __restrict__ scores, const int* __restrict__ topk,
            float* __restrict__ out) {
    __shared__ float red[1024];
    const int tid = threadIdx.x;
    float s = 0.0f;
    for (int t = tid; t < BB * KK; t += 1024) {
        int b = t / KK;
        int j = t - b * KK;
        int idx = topk[t];
        s += logf(scores[((size_t)b * KK + j) * NN + idx]);
    }
    red[tid] = s;
    __syncthreads();
    for (int off = 512; off > 0; off >>= 1) {
        if (tid < off) red[tid] += red[tid + off];
        __syncthreads();
    }
    if (tid == 0) out[0] = -red[0] / (float)(BB * KK);
}

// ---------------------------------------------------------------------------
extern "C" void kernel_launch(void* const* d_in, const int* in_sizes, int n_in,
                              void* d_out, int out_size, void* d_ws, size_t ws_size,
                              hipStream_t stream) {
    const float* scores = (const float*)d_in[0];  // [B, K, N]
    const float* src    = (const float*)d_in[1];  // [B, 3, N]
    const float* tgt    = (const float*)d_in[2];  // [B, 3, N]
    const float* rot    = (const float*)d_in[3];  // [B, 3, 3]
    const float* trans  = (const float*)d_in[4];  // [B, 3]
    float* out = (float*)d_out;

    // Workspace carve-up (all 16B aligned):
    char* ws = (char*)d_ws;
    size_t off = 0;
    float* srcT4  = (float*)(ws + off); off += (size_t)BB * NN * 4 * sizeof(float); // 512 KB
    float* tgtT4  = (float*)(ws + off); off += (size_t)BB * NN * 4 * sizeof(float); // 512 KB
    float* xx     = (float*)(ws + off); off += (size_t)BB * NN * sizeof(float);     // 128 KB
    float* nearst = (float*)(ws + off); off += (size_t)BB * NN * sizeof(float);     // 128 KB
    int*   topk   = (int*)  (ws + off); off += (size_t)BB * KK * sizeof(int);       //   8 KB

    // 1) transform + pack (BB*NN = 32768 points, exact grid)
    prep_kernel<<<(BB * NN) / 256, 256, 0, stream>>>(src, tgt, rot, trans,
                                                     srcT4, tgtT4, xx);
    // 2) WMMA nearest-neighbor distances: 2048 wave-tiles, 8 waves/block
    nn_kernel<<<(BB * (NN / 16)) / 8, 256, 0, stream>>>(srcT4, tgtT4, xx, nearst);
    // 3) per-batch stable full sort, keep first K indices
    topk_sort_kernel<<<BB, 1024, 0, stream>>>(nearst, topk);
    // 4) gather + reduce to scalar
    loss_kernel<<<1, 1024, 0, stream>>>(scores, topk, out);
}